// RelationPostProcessor_13615046329015
// MI455X (gfx1250) — compile-verified
//
#include <hip/hip_runtime.h>
#include <hip/hip_bf16.h>
#include <stdint.h>

typedef float v2f  __attribute__((ext_vector_type(2)));
typedef float v8f  __attribute__((ext_vector_type(8)));
typedef unsigned int u32x4 __attribute__((ext_vector_type(4)));
typedef int  i32x4 __attribute__((ext_vector_type(4)));
typedef int  i32x8 __attribute__((ext_vector_type(8)));

#define OBJ_CLS 151
#define REL_CLS 51
#define K2_RPB  128     // rel rows staged per block (8 waves x 16 rows)
#define SORT_EPB 2048   // elements per radix block (1 wave, 64 rounds)

// ---------------------------------------------------------------------------
// Kernel 1: object softmax -> bbox_scores + obj_pred  (one wave per row)
// ---------------------------------------------------------------------------
__global__ void obj_softmax_kernel(const float* __restrict__ logit,
                                   float* __restrict__ outPred,
                                   float* __restrict__ outScore,
                                   float* __restrict__ wsScore,
                                   int nBox)
{
    int wave = threadIdx.x >> 5;
    int lane = threadIdx.x & 31;
    int row  = blockIdx.x * 8 + wave;
    if (row >= nBox) return;
    const float* rp = logit + (long long)row * OBJ_CLS;

    float x[5];
    float m = -1e30f;
#pragma unroll
    for (int k = 0; k < 5; ++k) {
        int c = lane + k * 32;
        int cc = c < OBJ_CLS ? c : (OBJ_CLS - 1);   // safe clamped load
        float xv = rp[cc];
        x[k] = (c < OBJ_CLS) ? xv : -1e30f;
        m = fmaxf(m, x[k]);
    }
    for (int o = 16; o > 0; o >>= 1) m = fmaxf(m, __shfl_xor(m, o));

    float s = 0.f;
#pragma unroll
    for (int k = 0; k < 5; ++k) {
        int c = lane + k * 32;
        float e = __expf(x[k] - m);
        s += (c < OBJ_CLS) ? e : 0.f;
    }
    for (int o = 16; o > 0; o >>= 1) s += __shfl_xor(s, o);

    // foreground (cols 1..150) max + argmax; argmax of prob == argmax of logit
    float bv = -1e30f; int bc = 0x7fffffff;
#pragma unroll
    for (int k = 0; k < 5; ++k) {
        int c = lane + k * 32;
        bool ok = (c >= 1 && c < OBJ_CLS);
        bool better = ok && (x[k] > bv || (x[k] == bv && c < bc));
        bv = better ? x[k] : bv;
        bc = better ? c : bc;
    }
    for (int o = 16; o > 0; o >>= 1) {
        float pv = __shfl_xor(bv, o);
        int   pc = __shfl_xor(bc, o);
        bool better = (pv > bv || (pv == bv && pc < bc));
        bv = better ? pv : bv;
        bc = better ? pc : bc;
    }

    if (lane == 0) {
        float score = __expf(bv - m) / s;   // exp(fgmax-m)/sum = max fg prob
        outPred[row]  = (float)bc;          // argmax(fg)+1 == winning column
        outScore[row] = score;
        wsScore[row]  = score;
    }
}

// ---------------------------------------------------------------------------
// TDM: 1-D tile (elems f32 values) global -> LDS, single descriptor.
// clang-23 toolchain signature: 6 args (g0 u32x4, g1 i32x8, g2 i32x4,
// g3 i32x4, g4 i32x8, cpol i32).
// ---------------------------------------------------------------------------
__device__ __forceinline__ void tdm_load_1d(unsigned ldsAddr,
                                            unsigned long long gaddr,
                                            unsigned elems)
{
    u32x4 g0 = { 1u,                                     // count=1, user mode
                 ldsAddr,                                // lds_addr [63:32]
                 (unsigned)(gaddr & 0xffffffffull),      // global_addr lo
                 (unsigned)((gaddr >> 32) & 0x1ffffffull) | (2u << 30) }; // type=2
    i32x8 g1 = { (int)0x00020000u,                       // data_size=4B
                 (int)(elems << 16),                     // tensor_dim0 lo16
                 (int)((elems >> 16) | (1u << 16)),      // t_dim0 hi16 | t_dim1=1
                 (int)(elems << 16),                     // tile_dim0
                 (int)1,                                 // tile_dim1=1, tile_dim2=0
                 (int)elems,                             // t_dim0_stride lo32
                 (int)(elems << 16),                     // t_dim1_stride lo16
                 0 };
    i32x4 g2 = { 1, 1, (int)elems, 0 };                  // t_dim2=1, t_dim3=1, d2 stride
    i32x4 g3 = { (int)elems, (int)(1 << 16), 0, 0 };     // d3 stride, t_dim4=1
    i32x8 g4 = { 0, 0, 0, 0, 0, 0, 0, 0 };               // unused extra group
    __builtin_amdgcn_tensor_load_to_lds(g0, g1, g2, g3, g4, 0);
}

// ---------------------------------------------------------------------------
// Kernel 2: rel row stats. 8 waves/block, 16 rows per wave.
// TDM stages 128 rows into LDS; WMMA f32 16x16x4 (B = ones) computes row sums
// of exp(), co-executing with the v_exp_f32 stream.
// A-tile layout (32-bit, 16x4): lanes 0-15 hold K0,K1 of row=lane;
// lanes 16-31 hold K2,K3 of row=lane-16.
// Entire hot path is branch-free: clamped LDS loads + value selects, so no
// saveexec/branch pairs; out-of-range / tail-row lanes compute garbage that is
// confined to their own WMMA output rows and never stored.
// ---------------------------------------------------------------------------
__global__ void rel_stats_kernel(const float* __restrict__ relLogit,
                                 const long long* __restrict__ pairIdx,
                                 const float* __restrict__ bbox,
                                 float* __restrict__ rowMax,
                                 float* __restrict__ invSum,
                                 unsigned* __restrict__ relCls,
                                 unsigned* __restrict__ keys,
                                 unsigned* __restrict__ vals,
                                 int nRel)
{
    __shared__ float sh[K2_RPB * REL_CLS];
    __shared__ float shSums[8][16];

    const int       blockRow0 = blockIdx.x * K2_RPB;
    const long long elem0     = (long long)blockRow0 * REL_CLS;
    const long long total     = (long long)nRel * REL_CLS;
    long long rem = total - elem0;
    unsigned elems = (rem >= (long long)(K2_RPB * REL_CLS))
                       ? (unsigned)(K2_RPB * REL_CLS) : (unsigned)rem;

    if (threadIdx.x < 32) {   // wave 0 issues the DMA (EXEC ignored by TDM)
        unsigned ldsAddr = (unsigned)(unsigned long long)(const void*)&sh[0];
        unsigned long long ga =
            (unsigned long long)(const void*)relLogit + (unsigned long long)elem0 * 4ull;
        tdm_load_1d(ldsAddr, ga, elems);
        __builtin_amdgcn_s_wait_tensorcnt(0);
    }
    __syncthreads();

    const int wave = threadIdx.x >> 5;
    const int lane = threadIdx.x & 31;
    const int r    = lane & 15;        // row within the wave's 16-row group
    const int h    = lane >> 4;        // half: which K-pair this lane supplies
    const int lrow = wave * 16 + r;    // always < 128: LDS loads always safe
    const int row  = blockRow0 + lrow;
    const bool rowOK = row < nRel;
    const float* myRow = &sh[lrow * REL_CLS];

    // pass 1: load this lane's 26 columns once; row max / argmax / fg max.
    float xv[26];
    float m = -1e30f, bv = -1e30f, fgv = -1e30f;
    int bc = 0x7fffffff;
#pragma unroll
    for (int s = 0; s < 13; ++s) {
#pragma unroll
        for (int j = 0; j < 2; ++j) {
            int c  = s * 4 + h * 2 + j;
            int cc = c < (REL_CLS - 1) ? c : (REL_CLS - 1); // clamp (only s=12,h=1,j=1)
            float x = myRow[cc];
            bool ok = (c < REL_CLS);                         // folds except one case
            x = ok ? x : -1e30f;
            xv[s * 2 + j] = x;
            m = fmaxf(m, x);
            bool better = ok && (x > bv || (x == bv && c < bc));
            bv = better ? x : bv;
            bc = better ? c : bc;
            fgv = (c >= 1) ? fmaxf(fgv, x) : fgv;            // folds except c==0
        }
    }
    {   // combine the two halves of each row (partner = lane ^ 16)
        float pm = __shfl_xor(m, 16);   m = fmaxf(m, pm);
        float pv = __shfl_xor(bv, 16);  int pc = __shfl_xor(bc, 16);
        bool better = (pv > bv || (pv == bv && pc < bc));
        bv = better ? pv : bv;
        bc = better ? pc : bc;
        float pf = __shfl_xor(fgv, 16); fgv = fmaxf(fgv, pf);
    }

    // pass 2: 13 chained WMMAs accumulate exp row sums (B = ones).
    // xv[i] == -1e30 for the single out-of-range slot -> exp() == 0 exactly.
    // WMMA requires EXEC all-ones: no divergence anywhere in this loop.
    v8f acc = {};
    v2f ones2 = { 1.0f, 1.0f };
#pragma unroll
    for (int s = 0; s < 13; ++s) {
        v2f a = { __expf(xv[s * 2]     - m),
                  __expf(xv[s * 2 + 1] - m) };
        acc = __builtin_amdgcn_wmma_f32_16x16x4_f32(
                  false, a, false, ones2, (short)0, acc, false, false);
    }

    // D layout: VGPR j -> M=j (lanes 0-15) / M=8+j (lanes 16-31); all N equal.
    if (lane == 0) {
#pragma unroll
        for (int j = 0; j < 8; ++j) shSums[wave][j] = acc[j];
    }
    if (lane == 16) {
#pragma unroll
        for (int j = 0; j < 8; ++j) shSums[wave][8 + j] = acc[j];
    }
    __syncthreads();

    if (h == 0 && rowOK) {
        float sum = shSums[wave][r];
        float inv = 1.0f / sum;
        float relScore = __expf(fgv - m) * inv;          // max of prob[1:]
        long long si = pairIdx[2ll * row];
        long long oi = pairIdx[2ll * row + 1];
        float t = relScore * bbox[si] * bbox[oi];
        unsigned u   = __float_as_uint(t);
        unsigned asc = (u & 0x80000000u) ? ~u : (u | 0x80000000u);
        rowMax[row] = m;
        invSum[row] = inv;
        relCls[row] = (unsigned)bc;
        keys[row]   = ~asc;       // ascending radix on this = descending score
        vals[row]   = (unsigned)row;
    }
}

// ---------------------------------------------------------------------------
// Radix sort (4 x 8-bit LSD, stable). One wave per 2048-element tile.
// ---------------------------------------------------------------------------
__global__ void radix_hist_kernel(const unsigned* __restrict__ keysIn,
                                  unsigned* __restrict__ hist,
                                  int n, int numBlocks, int shift)
{
    __shared__ unsigned cnt[256];
    int lane = threadIdx.x;
    for (int d = lane; d < 256; d += 32) cnt[d] = 0;
    __syncthreads();
    int base = blockIdx.x * SORT_EPB;
    for (int i = 0; i < SORT_EPB / 32; ++i) {
        int idx = base + i * 32 + lane;
        if (idx < n) {
            unsigned d = (keysIn[idx] >> shift) & 255u;
            atomicAdd(&cnt[d], 1u);
        }
    }
    __syncthreads();
    for (int d = lane; d < 256; d += 32)
        hist[(long long)d * numBlocks + blockIdx.x] = cnt[d];
}

__global__ void radix_scan_kernel(unsigned* __restrict__ hist, int M)
{
    __shared__ unsigned s[256];
    int t = threadIdx.x;
    int chunk = (M + 255) / 256;
    int lo = t * chunk;
    int hi = lo + chunk; if (hi > M) hi = M;
    unsigned sum = 0;
    for (int i = lo; i < hi; ++i) sum += hist[i];
    s[t] = sum;
    __syncthreads();
    if (t == 0) {
        unsigned run = 0;
        for (int i = 0; i < 256; ++i) { unsigned v = s[i]; s[i] = run; run += v; }
    }
    __syncthreads();
    unsigned run = s[t];
    for (int i = lo; i < hi; ++i) { unsigned v = hist[i]; hist[i] = run; run += v; }
}

__global__ void radix_scatter_kernel(const unsigned* __restrict__ kin,
                                     const unsigned* __restrict__ vin,
                                     unsigned* __restrict__ kout,
                                     unsigned* __restrict__ vout,
                                     const unsigned* __restrict__ hist,
                                     int n, int numBlocks, int shift)
{
    __shared__ unsigned base[256];
    int lane = threadIdx.x;
    for (int d = lane; d < 256; d += 32)
        base[d] = hist[(long long)d * numBlocks + blockIdx.x];
    __syncthreads();

    int blk0 = blockIdx.x * SORT_EPB;
    unsigned ltmask = (1u << lane) - 1u;
    for (int i = 0; i < SORT_EPB / 32; ++i) {
        int idx = blk0 + i * 32 + lane;
        bool valid = idx < n;
        unsigned k = valid ? kin[idx] : 0xffffffffu;
        unsigned v = valid ? vin[idx] : 0u;
        unsigned d = (k >> shift) & 255u;

        // wave32 digit match via bit-serial ballots
        unsigned mask = 0xffffffffu;
#pragma unroll
        for (int b = 0; b < 8; ++b) {
            unsigned vote = (unsigned)__ballot((int)((d >> b) & 1u));
            mask &= ((d >> b) & 1u) ? vote : ~vote;
        }
        mask &= (unsigned)__ballot((int)valid);

        if (valid) {
            unsigned rank = __popc(mask & ltmask);
            unsigned dst  = base[d] + rank;
            kout[dst] = k;
            vout[dst] = v;
        }
        __syncthreads();
        if (valid && ((mask & ltmask) == 0u))   // group leader advances counter
            base[d] += __popc(mask);
        __syncthreads();
    }
}

// ---------------------------------------------------------------------------
// Kernel 4: gather outputs in sorted order (one wave per output row).
// Softmax probabilities recomputed from logits + (max, 1/sum) stats.
// ---------------------------------------------------------------------------
__global__ void gather_kernel(const float* __restrict__ relLogit,
                              const long long* __restrict__ pairIdx,
                              const unsigned* __restrict__ sortedIdx,
                              const float* __restrict__ rowMax,
                              const float* __restrict__ invSum,
                              const unsigned* __restrict__ relCls,
                              float* __restrict__ outPairs,
                              float* __restrict__ outCls,
                              float* __restrict__ outLbl,
                              int nRel)
{
    int wave = threadIdx.x >> 5;
    int lane = threadIdx.x & 31;
    int r = blockIdx.x * 8 + wave;
    if (r >= nRel) return;
    unsigned src = sortedIdx[r];
    const float* rp = relLogit + (long long)src * REL_CLS;
    __builtin_prefetch(rp, 0, 0);    // global_prefetch_b8 on the gathered row
    float m = rowMax[src], inv = invSum[src];
#pragma unroll
    for (int k = 0; k < 2; ++k) {
        int c = lane + k * 32;
        if (c < REL_CLS)
            outCls[(long long)r * REL_CLS + c] = __expf(rp[c] - m) * inv;
    }
    if (lane == 0) {
        outLbl[r] = (float)relCls[src];
        outPairs[2ll * r]     = (float)pairIdx[2ll * src];
        outPairs[2ll * r + 1] = (float)pairIdx[2ll * src + 1];
    }
}

// ---------------------------------------------------------------------------
extern "C" void kernel_launch(void* const* d_in, const int* in_sizes, int n_in,
                              void* d_out, int out_size, void* d_ws, size_t ws_size,
                              hipStream_t stream)
{
    const float*     objLogit = (const float*)d_in[0];
    const float*     relLogit = (const float*)d_in[1];
    const long long* pairIdx  = (const long long*)d_in[2];
    int nBox = in_sizes[0] / OBJ_CLS;
    int nRel = in_sizes[1] / REL_CLS;

    float* out      = (float*)d_out;
    float* outPred  = out;                                  // [nBox]
    float* outBbox  = outPred + nBox;                       // [nBox]
    float* outPairs = outBbox + nBox;                       // [nRel,2]
    float* outCls   = outPairs + 2ll * nRel;                // [nRel,51]
    float* outLbl   = outCls + (long long)nRel * REL_CLS;   // [nRel]

    char* ws = (char*)d_ws;
    size_t off = 0;
    auto carve = [&](size_t bytes) -> void* {
        void* p = ws + off;
        off = (off + bytes + 255) & ~(size_t)255;
        return p;
    };
    float*    wsBbox  = (float*)carve(sizeof(float) * nBox);
    float*    rowMax  = (float*)carve(sizeof(float) * (size_t)nRel);
    float*    invSum  = (float*)carve(sizeof(float) * (size_t)nRel);
    unsigned* relClsW = (unsigned*)carve(sizeof(unsigned) * (size_t)nRel);
    unsigned* keysA   = (unsigned*)carve(sizeof(unsigned) * (size_t)nRel);
    unsigned* valsA   = (unsigned*)carve(sizeof(unsigned) * (size_t)nRel);
    unsigned* keysB   = (unsigned*)carve(sizeof(unsigned) * (size_t)nRel);
    unsigned* valsB   = (unsigned*)carve(sizeof(unsigned) * (size_t)nRel);
    int numBlocks = (nRel + SORT_EPB - 1) / SORT_EPB;
    unsigned* hist = (unsigned*)carve(sizeof(unsigned) * 256 * (size_t)numBlocks);

    obj_softmax_kernel<<<(nBox + 7) / 8, 256, 0, stream>>>(
        objLogit, outPred, outBbox, wsBbox, nBox);

    rel_stats_kernel<<<(nRel + K2_RPB - 1) / K2_RPB, 256, 0, stream>>>(
        relLogit, pairIdx, wsBbox, rowMax, invSum, relClsW, keysA, valsA, nRel);

    unsigned *ki = keysA, *vi = valsA, *ko = keysB, *vo = valsB;
    for (int pass = 0; pass < 4; ++pass) {
        int shift = pass * 8;
        radix_hist_kernel<<<numBlocks, 32, 0, stream>>>(ki, hist, nRel, numBlocks, shift);
        radix_scan_kernel<<<1, 256, 0, stream>>>(hist, 256 * numBlocks);
        radix_scatter_kernel<<<numBlocks, 32, 0, stream>>>(ki, vi, ko, vo, hist,
                                                           nRel, numBlocks, shift);
        unsigned* t;
        t = ki; ki = ko; ko = t;
        t = vi; vi = vo; vo = t;
    }
    // after 4 passes, sorted keys/vals are back in the A buffers (== ki/vi)

    gather_kernel<<<(nRel + 7) / 8, 256, 0, stream>>>(
        relLogit, pairIdx, vi, rowMax, invSum, relClsW,
        outPairs, outCls, outLbl, nRel);
}